// GCNModel_20126216749771
// MI455X (gfx1250) — compile-verified
//
#include <hip/hip_runtime.h>

typedef __attribute__((ext_vector_type(2))) float v2f;
typedef __attribute__((ext_vector_type(8))) float v8f;

#define N_NODES 100000
#define N_EDGES 1600000
#define IN_F 128
#define HID 128

// ---------------------------------------------------------------------------
// 1) degree counting: one thread per edge, two f32 atomics
// ---------------------------------------------------------------------------
__global__ void degrees_kernel(const int* __restrict__ src, const int* __restrict__ dst,
                               float* __restrict__ deg_out, float* __restrict__ deg_in, int E) {
    int e = blockIdx.x * blockDim.x + threadIdx.x;
    if (e >= E) return;
    atomicAdd(&deg_out[src[e]], 1.0f);
    atomicAdd(&deg_in[dst[e]], 1.0f);
}

// deg -> deg^{-1/2} (0 for zero-degree), in place over both arrays (contiguous)
__global__ void ninv_kernel(float* __restrict__ deg, int n) {
    int i = blockIdx.x * blockDim.x + threadIdx.x;
    if (i >= n) return;
    float d = deg[i];
    deg[i] = (d > 0.0f) ? (1.0f / sqrtf(d)) : 0.0f;
}

// ---------------------------------------------------------------------------
// 2) x_scaled = in_feat * ninv_out[node]
// ---------------------------------------------------------------------------
__global__ void scale_kernel(const float* __restrict__ x, const float* __restrict__ ninv_out,
                             float* __restrict__ xs) {
    int i = blockIdx.x * blockDim.x + threadIdx.x;  // over N*128 exactly
    xs[i] = x[i] * ninv_out[i >> 7];
}

// ---------------------------------------------------------------------------
// 3) layer-1 edge aggregation: one wave32 per edge, float4 per lane
//    (wave reads a contiguous 512B row of x_scaled -> perfectly coalesced)
// ---------------------------------------------------------------------------
__global__ void agg128_kernel(const int* __restrict__ src, const int* __restrict__ dst,
                              const float* __restrict__ xs, float* __restrict__ agg, int E) {
    int e = (blockIdx.x * blockDim.x + threadIdx.x) >> 5;
    int lane = threadIdx.x & 31;
    if (e >= E) return;
    int s = src[e];
    int d = dst[e];
    const float4 v = *(const float4*)(xs + (size_t)s * 128 + lane * 4);
    float* dp = agg + (size_t)d * 128 + lane * 4;
    atomicAdd(dp + 0, v.x);
    atomicAdd(dp + 1, v.y);
    atomicAdd(dp + 2, v.z);
    atomicAdd(dp + 3, v.w);
}

// ---------------------------------------------------------------------------
// 4) h1 = relu( (agg @ W1) * ninv_in + b1 )   -- fp32 WMMA 16x16x4
//    block = 256 threads = 8 waves; block owns a 16-row slice, wave w owns
//    columns [16w,16w+16). A tile (16x128) staged through LDS.
// ---------------------------------------------------------------------------
__global__ void gemm1_relu_kernel(const float* __restrict__ agg, const float* __restrict__ W1,
                                  const float* __restrict__ b1, const float* __restrict__ ninv_in,
                                  float* __restrict__ h1) {
    __shared__ float As[16 * 128];  // 8 KB

    const int tid  = threadIdx.x;
    const int lane = tid & 31;
    const int wave = tid >> 5;
    const int row0 = blockIdx.x * 16;

    // cooperative A-tile load: 512 float4s, 2 per thread
    const float4* g4 = (const float4*)(agg + (size_t)row0 * 128);
    float4* s4 = (float4*)As;
    s4[tid * 2 + 0] = g4[tid * 2 + 0];
    s4[tid * 2 + 1] = g4[tid * 2 + 1];
    __syncthreads();

    const int col0 = wave * 16;
    const int mn   = lane & 15;          // A row within tile == B column within tile
    const int kb   = (lane >> 4) * 2;    // lanes 16..31 hold K+2,K+3

    v8f c = {};
#pragma unroll
    for (int kk = 0; kk < 32; ++kk) {
        const int k = kk * 4 + kb;
        v2f a, b;
        a.x = As[mn * 128 + k];
        a.y = As[mn * 128 + k + 1];
        b.x = W1[(size_t)k * 128 + col0 + mn];
        b.y = W1[(size_t)(k + 1) * 128 + col0 + mn];
        // D = A(16x4,f32) * B(4x16,f32) + C  ->  v_wmma_f32_16x16x4_f32
        c = __builtin_amdgcn_wmma_f32_16x16x4_f32(false, a, false, b,
                                                  (short)0, c, false, false);
    }

    // C/D layout: VGPR r, lanes 0-15 -> M=r ; lanes 16-31 -> M=r+8 ; N=lane&15
    const int n     = col0 + mn;
    const int mbase = row0 + ((lane >> 4) ? 8 : 0);
    const float bn  = b1[n];
#pragma unroll
    for (int r = 0; r < 8; ++r) {
        const int m = mbase + r;
        float v = c[r] * ninv_in[m] + bn;
        h1[(size_t)m * 128 + n] = fmaxf(v, 0.0f);
    }
}

// ---------------------------------------------------------------------------
// 5) layer-2 per-node matvec: y[n][0:2] = (h1[n]*ninv_out[n]) @ W2  (128x2)
//    one wave per node, shuffle reduction
// ---------------------------------------------------------------------------
__global__ void l2_matvec_kernel(const float* __restrict__ h1, const float* __restrict__ ninv_out,
                                 const float* __restrict__ W2, float* __restrict__ y_node, int N) {
    int node = blockIdx.x * 8 + (threadIdx.x >> 5);
    int lane = threadIdx.x & 31;
    if (node >= N) return;
    const float s = ninv_out[node];
    float y0 = 0.0f, y1 = 0.0f;
#pragma unroll
    for (int j = 0; j < 4; ++j) {
        int f = lane + j * 32;
        float v = h1[(size_t)node * 128 + f] * s;
        y0 += v * W2[f * 2 + 0];
        y1 += v * W2[f * 2 + 1];
    }
#pragma unroll
    for (int off = 16; off > 0; off >>= 1) {
        y0 += __shfl_xor(y0, off, 32);
        y1 += __shfl_xor(y1, off, 32);
    }
    if (lane == 0) {
        y_node[node * 2 + 0] = y0;
        y_node[node * 2 + 1] = y1;
    }
}

// ---------------------------------------------------------------------------
// 6) layer-2 edge aggregation (2 floats per edge)
// ---------------------------------------------------------------------------
__global__ void agg2_kernel(const int* __restrict__ src, const int* __restrict__ dst,
                            const float* __restrict__ y_node, float* __restrict__ out_acc, int E) {
    int e = blockIdx.x * blockDim.x + threadIdx.x;
    if (e >= E) return;
    int s = src[e];
    int d = dst[e];
    atomicAdd(&out_acc[d * 2 + 0], y_node[s * 2 + 0]);
    atomicAdd(&out_acc[d * 2 + 1], y_node[s * 2 + 1]);
}

// ---------------------------------------------------------------------------
// 7) out = out_acc * ninv_in + b2
// ---------------------------------------------------------------------------
__global__ void final_kernel(const float* __restrict__ out_acc, const float* __restrict__ ninv_in,
                             const float* __restrict__ b2, float* __restrict__ out, int N) {
    int n = blockIdx.x * blockDim.x + threadIdx.x;
    if (n >= N) return;
    float ni = ninv_in[n];
    out[n * 2 + 0] = out_acc[n * 2 + 0] * ni + b2[0];
    out[n * 2 + 1] = out_acc[n * 2 + 1] * ni + b2[1];
}

extern "C" void kernel_launch(void* const* d_in, const int* in_sizes, int n_in,
                              void* d_out, int out_size, void* d_ws, size_t ws_size,
                              hipStream_t stream) {
    const float* in_feat = (const float*)d_in[0];
    const float* W1      = (const float*)d_in[1];
    const float* b1      = (const float*)d_in[2];
    const float* W2      = (const float*)d_in[3];
    const float* b2      = (const float*)d_in[4];
    const int*   src     = (const int*)d_in[5];
    const int*   dst     = (const int*)d_in[6];
    float*       out     = (float*)d_out;

    const int N = N_NODES;
    const int E = N_EDGES;

    // workspace layout (floats): ~105 MB total
    float* ws      = (float*)d_ws;
    float* deg_out = ws;                          // N   (becomes ninv_out)
    float* deg_in  = ws + N;                      // N   (becomes ninv_in)
    float* xs      = ws + 2 * (size_t)N;          // N*128 (x_scaled, reused as h1)
    float* agg     = xs + (size_t)N * 128;        // N*128
    float* y_node  = agg + (size_t)N * 128;       // 2N
    float* out_acc = y_node + 2 * (size_t)N;      // 2N

    hipMemsetAsync(deg_out, 0, 2 * (size_t)N * sizeof(float), stream);
    hipMemsetAsync(agg, 0, (size_t)N * 128 * sizeof(float), stream);
    hipMemsetAsync(out_acc, 0, 2 * (size_t)N * sizeof(float), stream);

    degrees_kernel<<<(E + 255) / 256, 256, 0, stream>>>(src, dst, deg_out, deg_in, E);
    ninv_kernel<<<(2 * N + 255) / 256, 256, 0, stream>>>(deg_out, 2 * N);
    scale_kernel<<<(N * 128) / 256, 256, 0, stream>>>(in_feat, deg_out, xs);
    agg128_kernel<<<E / 8, 256, 0, stream>>>(src, dst, xs, agg, E);           // wave/edge
    gemm1_relu_kernel<<<N / 16, 256, 0, stream>>>(agg, W1, b1, deg_in, xs);   // h1 -> xs
    l2_matvec_kernel<<<N / 8, 256, 0, stream>>>(xs, deg_out, W2, y_node, N);
    agg2_kernel<<<(E + 255) / 256, 256, 0, stream>>>(src, dst, y_node, out_acc, E);
    final_kernel<<<(N + 255) / 256, 256, 0, stream>>>(out_acc, deg_in, b2, out, N);
}